// AxialAttention_49718541418542
// MI455X (gfx1250) — compile-verified
//
#include <hip/hip_runtime.h>

typedef __attribute__((ext_vector_type(2))) float v2f;
typedef __attribute__((ext_vector_type(8))) float v8f;
typedef __attribute__((ext_vector_type(4))) unsigned int u32x4;
typedef __attribute__((ext_vector_type(4))) int i32x4;
typedef __attribute__((ext_vector_type(8))) int i32x8;

#define KS    48
#define CIN   64
#define SX    66          /* s_x row stride (64 data + 2 pad dwords, 8B-aligned rows) */
#define SQ    50          /* s_qkv row stride */
#define SR    50          /* qrT/krT row stride */
#define SS    50          /* sim row stride */
#define NWAVE 4
#define HSTR  (48*48*64)
#define WSTR  (48*64)

#ifndef __has_builtin
#define __has_builtin(x) 0
#endif
#if __has_builtin(__builtin_amdgcn_tensor_load_to_lds) && __has_builtin(__builtin_amdgcn_s_wait_tensorcnt)
#define HAVE_TDM 1
#warning "CDNA5 probe: tensor_load_to_lds path ENABLED"
#else
#define HAVE_TDM 0
#warning "CDNA5 probe: tensor_load_to_lds NOT available - manual fallback in use"
#endif

static __device__ __forceinline__ v8f wmma_f32(v2f a, v2f b, v8f c) {
    return __builtin_amdgcn_wmma_f32_16x16x4_f32(false, a, false, b, (short)0, c, false, false);
}

__global__ __launch_bounds__(128) void axial_kernel(
    const float* __restrict__ xg,   const float* __restrict__ wqkv,
    const float* __restrict__ relg,
    const float* __restrict__ gqkv, const float* __restrict__ bqkv,
    const float* __restrict__ gsim, const float* __restrict__ bsim,
    const float* __restrict__ gout, const float* __restrict__ bout,
    float* __restrict__ outg)
{
    __shared__ float s_x[KS * SX];             // [h][c] (TDM-padded rows)
    __shared__ float s_qkv[128 * SQ];          // [o][h]  (BN applied)
    __shared__ float s_qrT[NWAVE][KS * SR];    // qrT[j][i] = QR[i][j]  (fragment-major)
    __shared__ float s_krT[NWAVE][KS * SR];    // krT[j][i] = KR[i][j]
    __shared__ float s_sim[NWAVE][16 * SS];
    __shared__ float s_rel[16 * 95];

    const int tid  = threadIdx.x;
    const int wv   = tid >> 5;
    const int lane = tid & 31;
    const int half = lane >> 4;
    const int rA   = lane & 15;
    const int kb   = half << 1;                // K offset for A/B fragments
    const int bw   = blockIdx.x / KS;
    const int bl   = blockIdx.x % KS;
    const float inv_s = rsqrtf(1.001f);        // 1/sqrt(1+eps)

    // ---------- phase 0: stage X pencil (48 rows x 64 floats, row stride HSTR) ----------
    const float* xp = xg + bw * WSTR + bl * CIN;
#if HAVE_TDM
    if (wv == 0) {
        unsigned lds_addr = (unsigned)(size_t)(void*)&s_x[0];
        unsigned long long ga = (unsigned long long)(size_t)(const void*)xp;
        u32x4 g0;
        g0[0] = 1u;                                            // count=1, load descriptor
        g0[1] = lds_addr;                                      // lds_addr
        g0[2] = (unsigned)(ga & 0xffffffffu);                  // global_addr[31:0]
        g0[3] = (unsigned)((ga >> 32) & 0x01ffffffu) | (2u << 30); // global_addr[56:32] | type=2
        i32x8 g1;
        g1[0] = (int)((2u << 16) | (1u << 20) | (5u << 22) | (1u << 25));
                 // data_size=4B, pad_enable, pad_interval=64dw, pad_amount=2dw
        g1[1] = (int)(64u << 16);                              // tensor_dim0 = 64
        g1[2] = (int)(48u << 16);                              // tensor_dim1 = 48
        g1[3] = (int)(64u << 16);                              // tile_dim0 = 64
        g1[4] = 48;                                            // tile_dim1 = 48 (tile_dim2 = 0)
        g1[5] = (int)(unsigned)HSTR;                           // tensor_dim0_stride = 147456
        g1[6] = 0;
        g1[7] = 0;
        i32x4 g2 = {0, 0, 0, 0};
        i32x4 g3 = {0, 0, 0, 0};
#if __clang_major__ >= 23
        i32x8 g4 = {0, 0, 0, 0, 0, 0, 0, 0};
        __builtin_amdgcn_tensor_load_to_lds(g0, g1, g2, g3, g4, 0);
#else
        __builtin_amdgcn_tensor_load_to_lds(g0, g1, g2, g3, 0);
#endif
    }
#endif
    for (int idx = tid; idx < 16 * 95; idx += 128) s_rel[idx] = relg[idx];
#if HAVE_TDM
    if (wv == 0) __builtin_amdgcn_s_wait_tensorcnt(0);
#else
    for (int idx = tid; idx < KS * CIN; idx += 128) {
        int h = idx >> 6, c = idx & 63;
        s_x[h * SX + c] = xp[h * HSTR + c];
    }
#endif
    __syncthreads();

    // ---------- phase 1: qkv = W(128x64) @ X(64x48), BN along h ----------
    {
        const float sq0 = gqkv[rA] * inv_s,      bq0 = bqkv[rA];
        const float sq1 = gqkv[16 + rA] * inv_s, bq1 = bqkv[16 + rA];
        const float sq2 = gqkv[32 + rA] * inv_s, bq2 = bqkv[32 + rA];
        for (int mt = 2 * wv; mt < 2 * wv + 2; ++mt) {
            v8f acc0 = {}, acc1 = {}, acc2 = {};
            const float* wrow = wqkv + (16 * mt + rA) * CIN;
            for (int kc = 0; kc < 16; ++kc) {
                int k0 = 4 * kc + kb;
                v2f a = *(const v2f*)(wrow + k0);
                acc0 = wmma_f32(a, *(const v2f*)&s_x[rA * SX + k0],        acc0);
                acc1 = wmma_f32(a, *(const v2f*)&s_x[(16 + rA) * SX + k0], acc1);
                acc2 = wmma_f32(a, *(const v2f*)&s_x[(32 + rA) * SX + k0], acc2);
            }
            #pragma unroll
            for (int r = 0; r < 8; ++r) {
                int o = 16 * mt + r + 8 * half;
                s_qkv[o * SQ + rA]      = acc0[r] * sq0 + bq0;
                s_qkv[o * SQ + 16 + rA] = acc1[r] * sq1 + bq1;
                s_qkv[o * SQ + 32 + rA] = acc2[r] * sq2 + bq2;
            }
        }
    }
    __syncthreads();

    // ---------- phase 2: per-group attention (wave-private LDS) ----------
    float* qrT = s_qrT[wv];
    float* krT = s_krT[wv];
    float* sim = s_sim[wv];

    for (int gi = 0; gi < 2; ++gi) {
        const int g = wv + NWAVE * gi;
        const float* qrow = s_qkv + (g * 16) * SQ;   // rows 0-3 q, 4-7 k, 8-15 v

        // 2a: qrT[a][b] = QR[b][a] = sum_c q[c][b]*rel[c][b-a+47]
        //     krT[a][b] = KR[b][a] = sum_c k[c][a]*rel[4+c][a-b+47]
        for (int e = 2 * lane; e < KS * KS; e += 64) {
            int a = e / KS, b = e - a * KS;          // b even
            float q0 = 0.f, q1 = 0.f, k0v = 0.f, k1v = 0.f;
            #pragma unroll
            for (int c = 0; c < 4; ++c) {
                const float* rq = s_rel + c * 95 + 47 - a;
                v2f qp = *(const v2f*)(qrow + c * SQ + b);
                q0 += qp.x * rq[b];
                q1 += qp.y * rq[b + 1];
                const float* rk = s_rel + (4 + c) * 95 + 47 + a;
                float kc = qrow[(4 + c) * SQ + a];
                k0v += kc * rk[-b];
                k1v += kc * rk[-(b + 1)];
            }
            v2f vq; vq.x = q0;  vq.y = q1;
            v2f vk; vk.x = k0v; vk.y = k1v;
            *(v2f*)&qrT[a * SR + b] = vq;
            *(v2f*)&krT[a * SR + b] = vk;
        }

        for (int ti = 0; ti < 3; ++ti) {
            const int ibase = 16 * ti;
            // qk[i,j] = sum_m QR[m][i]*KR[m][j]; A row i from qrT[i][m], B from krT[j][m]
            v8f acc[3] = {};
            for (int mc = 0; mc < 12; ++mc) {
                int m0 = 4 * mc + kb;
                v2f a = *(const v2f*)&qrT[(ibase + rA) * SR + m0];
                #pragma unroll
                for (int tj = 0; tj < 3; ++tj) {
                    v2f b = *(const v2f*)&krT[(16 * tj + rA) * SR + m0];
                    acc[tj] = wmma_f32(a, b, acc[tj]);
                }
            }
            // logits = ssim_j*(qk + QR[i][j] + KR[i][j]) + 3*bsim_j (D-fragment layout)
            #pragma unroll
            for (int tj = 0; tj < 3; ++tj) {
                int j = 16 * tj + rA;
                float ss = gsim[j] * inv_s, b3 = 3.0f * bsim[j];
                #pragma unroll
                for (int r = 0; r < 8; r += 2) {
                    int i = ibase + r + 8 * half;             // even -> 8B-aligned pair
                    v2f qv = *(const v2f*)&qrT[j * SR + i];
                    v2f kv = *(const v2f*)&krT[j * SR + i];
                    acc[tj][r]     = ss * (acc[tj][r]     + qv.x + kv.x) + b3;
                    acc[tj][r + 1] = ss * (acc[tj][r + 1] + qv.y + kv.y) + b3;
                }
            }
            // row softmax over j (48 values: 3 regs x 16 lanes of this half)
            #pragma unroll
            for (int r = 0; r < 8; ++r) {
                float m = fmaxf(fmaxf(acc[0][r], acc[1][r]), acc[2][r]);
                for (int d = 1; d < 16; d <<= 1) m = fmaxf(m, __shfl_xor(m, d, 32));
                float s = 0.f;
                #pragma unroll
                for (int tj = 0; tj < 3; ++tj) { float e = __expf(acc[tj][r] - m); acc[tj][r] = e; s += e; }
                for (int d = 1; d < 16; d <<= 1) s += __shfl_xor(s, d, 32);
                float is = 1.0f / s;
                int il = r + 8 * half;
                #pragma unroll
                for (int tj = 0; tj < 3; ++tj) sim[il * SS + 16 * tj + rA] = acc[tj][r] * is;
            }
            // sv[c,i] = sum_j sim[i,j]*v[c,j]  (A = sim tile, B = v^T, N padded 8->16)
            v8f sacc = {};
            const int vc = rA & 7;
            for (int jc = 0; jc < 12; ++jc) {
                int j0 = 4 * jc + kb;
                v2f a = *(const v2f*)&sim[rA * SS + j0];
                v2f b = *(const v2f*)&s_qkv[(g * 16 + 8 + vc) * SQ + j0];
                if (rA >= 8) { b.x = 0.0f; b.y = 0.0f; }
                sacc = wmma_f32(a, b, sacc);
            }
            // sve[c,i] = sum_j sim[i,j]*rel[8+c][i-j+47]
            // split 8 r-values across the two 8-lane subgroups; exchange via shfl_xor(8)
            const int rbase = (rA >> 3) << 2;    // lanes 0-7 -> r 0..3, lanes 8-15 -> r 4..7
            float svev[4];
            #pragma unroll
            for (int rr = 0; rr < 4; ++rr) {
                int r  = rbase + rr;
                int il = r + 8 * half;
                int i  = ibase + il;
                const float* vrel = s_rel + (8 + vc) * 95 + (i + 47);
                const float* srow = sim + il * SS;
                float sve = 0.f;
                for (int j = 0; j < KS; j += 2) {
                    v2f sp = *(const v2f*)(srow + j);
                    sve += sp.x * vrel[-j];
                    sve += sp.y * vrel[-(j + 1)];
                }
                svev[rr] = sve;
            }
            float svoth[4];
            #pragma unroll
            for (int rr = 0; rr < 4; ++rr) svoth[rr] = __shfl_xor(svev[rr], 8, 32);
            if (rA < 8) {
                #pragma unroll
                for (int r = 0; r < 8; ++r) {
                    int i = ibase + r + 8 * half;
                    float sve = (r < 4) ? svev[r] : svoth[r - 4];
                    float so = gout[i] * inv_s * (sacc[r] + sve) + 2.0f * bout[i];
                    outg[i * HSTR + bw * WSTR + bl * CIN + g * 8 + rA] = so;
                }
            }
        }
    }
}

extern "C" void kernel_launch(void* const* d_in, const int* in_sizes, int n_in,
                              void* d_out, int out_size, void* d_ws, size_t ws_size,
                              hipStream_t stream) {
    const float* x    = (const float*)d_in[0];
    const float* wqkv = (const float*)d_in[1];
    const float* rel  = (const float*)d_in[2];
    const float* gq   = (const float*)d_in[3];
    const float* bq   = (const float*)d_in[4];
    const float* gs   = (const float*)d_in[5];
    const float* bs   = (const float*)d_in[6];
    const float* go   = (const float*)d_in[7];
    const float* bo   = (const float*)d_in[8];
    float* out        = (float*)d_out;
    axial_kernel<<<dim3(48 * 48), dim3(128), 0, stream>>>(
        x, wqkv, rel, gq, bq, gs, bs, go, bo, out);
}